// boolean_product_8684423873129
// MI455X (gfx1250) — compile-verified
//
#include <hip/hip_runtime.h>
#include <hip/hip_bf16.h>
#include <stdint.h>

#define B_   2
#define N_   2048
#define D_   128
#define K_   16
#define ROWS (B_ * N_)          // 4096

typedef float v2f  __attribute__((ext_vector_type(2)));
typedef float v8f  __attribute__((ext_vector_type(8)));
typedef unsigned int u32x4 __attribute__((ext_vector_type(4)));
typedef int   i32x4 __attribute__((ext_vector_type(4)));
typedef int   i32x8 __attribute__((ext_vector_type(8)));

__device__ __forceinline__ v8f wmma_f32(v2f a, v2f b, v8f c) {
  // V_WMMA_F32_16X16X4_F32 : D = A(16x4) * B(4x16) + C(16x16), all fp32
  return __builtin_amdgcn_wmma_f32_16x16x4_f32(
      /*neg_a=*/false, a, /*neg_b=*/false, b,
      /*c_mod=*/(short)0, c, /*reuse_a=*/false, /*reuse_b=*/false);
}

// ---------------------------------------------------------------------------
// 1) xe = x @ W   (one wave -> 16 rows x 128 cols; 8 accumulators, K step 4)
// ---------------------------------------------------------------------------
__global__ void embed_kernel(const float* __restrict__ x,
                             const float* __restrict__ W,
                             float* __restrict__ xe,
                             float* __restrict__ xout) {
  const int lane = threadIdx.x & 31;
  const int wave = threadIdx.x >> 5;
  const int tile = blockIdx.x * 4 + wave;     // 256 tiles of 16 rows
  const int row0 = tile * 16;
  const int mrow = lane & 15;
  const int koff = (lane >> 4) << 1;          // lanes 16..31 hold K+2,K+3

  v8f acc[8];
  const v8f vzero = {0.f,0.f,0.f,0.f,0.f,0.f,0.f,0.f};
#pragma unroll
  for (int j = 0; j < 8; ++j) acc[j] = vzero;

  for (int kk = 0; kk < D_; kk += 4) {
    v2f a;
    const float* xr = x + (size_t)(row0 + mrow) * D_ + kk + koff;
    a.x = xr[0]; a.y = xr[1];
#pragma unroll
    for (int j = 0; j < 8; ++j) {
      v2f bb;
      const float* wr = W + (size_t)(kk + koff) * D_ + j * 16 + mrow;
      bb.x = wr[0];
      bb.y = wr[D_];
      acc[j] = wmma_f32(a, bb, acc[j]);
    }
  }

  const int hi = lane >> 4;
#pragma unroll
  for (int j = 0; j < 8; ++j) {
#pragma unroll
    for (int r = 0; r < 8; ++r) {
      const int row = row0 + hi * 8 + r;
      const int col = j * 16 + mrow;
      const float v = acc[j][r];
      xe[(size_t)row * D_ + col]   = v;
      xout[(size_t)row * D_ + col] = v;
    }
  }
}

// ---------------------------------------------------------------------------
// 2) x2[row] = sum_d xe[row][d]^2   (one wave per row)
// ---------------------------------------------------------------------------
__global__ void rownorm_kernel(const float* __restrict__ xe,
                               float* __restrict__ x2) {
  const int wave = (int)((blockIdx.x * blockDim.x + threadIdx.x) >> 5);
  const int lane = threadIdx.x & 31;
  if (wave >= ROWS) return;
  const float* r = xe + (size_t)wave * D_;
  float s = 0.f;
  for (int d = lane; d < D_; d += 32) { const float v = r[d]; s += v * v; }
  for (int off = 16; off; off >>= 1) s += __shfl_down(s, off, 32);
  if (lane == 0) x2[wave] = s;
}

// ---------------------------------------------------------------------------
// 3) count[b][c] = sum_m A_init[b][m][c]
// ---------------------------------------------------------------------------
__global__ void colcount_kernel(const int* __restrict__ A,
                                float* __restrict__ count) {
  const int t = blockIdx.x * blockDim.x + threadIdx.x;  // 0..4095
  const int b = t / N_, c = t % N_;
  const int* g = A + (size_t)b * N_ * N_ + c;
  int s = 0;
  for (int m = 0; m < N_; ++m) s += g[(size_t)m * N_];
  count[t] = (float)s;
}

// ---------------------------------------------------------------------------
// 4) prob = exp(-max(||xi||^2+||xj||^2-2 xi.xj,0) * exp(clip(T)))
//    One wave -> 16 x 128 strip (8 WMMA tiles). A-tile (16x128) staged into
//    LDS via the Tensor Data Mover and reused across all 256 WMMAs.
// ---------------------------------------------------------------------------
__global__ void gram_prob_kernel(const float* __restrict__ xe,
                                 const float* __restrict__ x2,
                                 const float* __restrict__ temp,
                                 float* __restrict__ prob) {
  __shared__ float smem[4 * 16 * D_];           // 4 waves x 8KB
  const int lane = threadIdx.x & 31;
  const int wave = threadIdx.x >> 5;
  const int tile = blockIdx.x * 4 + wave;       // B*128*16 = 4096 strips
  const int b    = tile >> 11;                  // / (128*16)
  const int rem  = tile & 2047;
  const int n0   = (rem >> 4) * 16;             // row tile
  const int m0   = (rem & 15) * 128;            // column strip
  float* myLds = smem + wave * (16 * D_);

  // --- TDM descriptor: 2D tile, 128 elems x 16 rows, 4B elements ---
  const float* src = xe + (size_t)(b * N_ + n0) * D_;
  const unsigned long long ga = (unsigned long long)(uintptr_t)src;
  u32x4 g0;
  g0.x = 1u;                                           // count=1, user mode
  g0.y = (unsigned)(uintptr_t)myLds;                   // LDS byte address
  g0.z = (unsigned)(ga & 0xFFFFFFFFull);               // global_addr[31:0]
  g0.w = (unsigned)((ga >> 32) & 0x01FFFFFFull) | (2u << 30);  // addr[56:32], type=2
  i32x8 g1;
  g1[0] = 0x00020000;              // data_size = 2 (4 bytes), no multicast
  g1[1] = (D_ << 16);              // tensor_dim0 = 128
  g1[2] = ((ROWS & 0xFFFF) << 16); // tensor_dim1 low16
  g1[3] = (D_ << 16);              // tile_dim0 = 128
  g1[4] = 16;                      // tile_dim1 = 16, tile_dim2 = 0
  g1[5] = D_;                      // tensor_dim0_stride = 128
  g1[6] = 0;
  g1[7] = 0;
  const i32x4 z4 = {0, 0, 0, 0};
  const i32x8 z8 = {0, 0, 0, 0, 0, 0, 0, 0};
  __builtin_amdgcn_tensor_load_to_lds(g0, g1, z4, z4, z8, 0);
  __builtin_amdgcn_s_wait_tensorcnt(0);
  __builtin_amdgcn_wave_barrier();

  const int mrow = lane & 15;
  const int koff = (lane >> 4) << 1;

  v8f acc[8];
  const v8f vzero = {0.f,0.f,0.f,0.f,0.f,0.f,0.f,0.f};
#pragma unroll
  for (int j = 0; j < 8; ++j) acc[j] = vzero;

  const float* bbase = xe + (size_t)(b * N_ + m0 + mrow) * D_;  // j advances by 16 rows
  for (int kk = 0; kk < D_; kk += 4) {
    v2f a;
    a.x = myLds[mrow * D_ + kk + koff];
    a.y = myLds[mrow * D_ + kk + koff + 1];
#pragma unroll
    for (int j = 0; j < 8; ++j) {
      v2f bb;
      const float* br = bbase + (size_t)(j * 16) * D_ + kk + koff;
      bb.x = br[0];
      bb.y = br[1];
      acc[j] = wmma_f32(a, bb, acc[j]);
    }
  }

  const float T   = temp[0];
  const float tsc = __expf(fminf(fmaxf(T, -5.f), 5.f));
  const int hi    = lane >> 4;
  float x2n[8];
#pragma unroll
  for (int r = 0; r < 8; ++r) x2n[r] = x2[b * N_ + n0 + hi * 8 + r];
#pragma unroll
  for (int j = 0; j < 8; ++j) {
    const int   col = m0 + j * 16 + mrow;
    const float x2m = x2[b * N_ + col];
#pragma unroll
    for (int r = 0; r < 8; ++r) {
      const int n = n0 + hi * 8 + r;
      float d2 = fmaxf(x2n[r] + x2m - 2.f * acc[j][r], 0.f);
      prob[((size_t)(b * N_) + n) * N_ + col] = __expf(-d2 * tsc);
    }
  }
}

// ---------------------------------------------------------------------------
// 5) m1 = log(max(prob@g,1e-24)) - log(max(count,1e-8))   (the 34 GFLOP GEMM)
//    One wave -> 16 x 128 strip: A (prob) kept in registers for 8 WMMAs.
// ---------------------------------------------------------------------------
__global__ void m1_gemm_kernel(const float* __restrict__ prob,
                               const int* __restrict__ A,
                               const float* __restrict__ count,
                               float* __restrict__ m1) {
  const int lane = threadIdx.x & 31;
  const int wave = threadIdx.x >> 5;
  const int tile = blockIdx.x * 8 + wave;       // B*128*16 = 4096 strips
  const int b    = tile >> 11;
  const int rem  = tile & 2047;
  const int n0   = (rem >> 4) * 16;
  const int c0   = (rem & 15) * 128;
  const int mrow = lane & 15;
  const int koff = (lane >> 4) << 1;
  const float* pb = prob + (size_t)b * N_ * N_;
  const int*   gb = A    + (size_t)b * N_ * N_;

  v8f acc[8];
  const v8f vzero = {0.f,0.f,0.f,0.f,0.f,0.f,0.f,0.f};
#pragma unroll
  for (int j = 0; j < 8; ++j) acc[j] = vzero;

  const float* arow = pb + (size_t)(n0 + mrow) * N_;
  for (int kk = 0; kk < N_; kk += 4) {
    v2f a;
    a.x = arow[kk + koff];
    a.y = arow[kk + koff + 1];
    const int* gr0 = gb + (size_t)(kk + koff) * N_ + c0 + mrow;     // coalesced
    const int* gr1 = gr0 + N_;
#pragma unroll
    for (int j = 0; j < 8; ++j) {
      v2f bb;
      bb.x = (float)gr0[j * 16];
      bb.y = (float)gr1[j * 16];
      acc[j] = wmma_f32(a, bb, acc[j]);
    }
  }

  const int hi = lane >> 4;
#pragma unroll
  for (int j = 0; j < 8; ++j) {
    const int   col = c0 + j * 16 + mrow;
    const float lc  = __logf(fmaxf(count[b * N_ + col], 1e-8f));
#pragma unroll
    for (int r = 0; r < 8; ++r) {
      const int n = n0 + hi * 8 + r;
      m1[((size_t)(b * N_) + n) * N_ + col] =
          __logf(fmaxf(acc[j][r], 1e-24f)) - lc;
    }
  }
}

// ---------------------------------------------------------------------------
// 6) merge + Gumbel perturb + per-row top-16 (one wave per row)
// ---------------------------------------------------------------------------
__global__ void merge_topk_kernel(const float* __restrict__ m1,
                                  const float* __restrict__ q,
                                  float* __restrict__ edges,
                                  float* __restrict__ lprobs) {
  __shared__ float buf[2][N_];                  // 16KB
  const int lane = threadIdx.x & 31;
  const int wave = threadIdx.x >> 5;
  const int row  = blockIdx.x * 2 + wave;       // 0..4095
  const int b    = row / N_;
  const int n    = row % N_;
  const float* m1b  = m1 + (size_t)b * N_ * N_;
  const float* qrow = q  + (size_t)row * N_;
  float* rb = buf[wave];

  for (int m = lane; m < N_; m += 32) {
    const float a  = m1b[(size_t)n * N_ + m];
    const float bt = m1b[(size_t)m * N_ + n];
    const float mx = fmaxf(a, bt), mn = fminf(a, bt);
    const float la = mx + log1pf(__expf(mn - mx));        // logaddexp
    rb[m] = (la - 0.69314718056f) + __logf(-__logf(qrow[m] + 1e-8f));
  }
  __syncthreads();

  float* e0 = edges;                         // edges[0]: B*N*K
  float* e1 = edges + (size_t)B_ * N_ * K_;  // edges[1]: B*N*K
  float* lp = lprobs + (size_t)row * K_;
  const size_t eb = (size_t)row * K_;

  for (int k = 0; k < K_; ++k) {
    float best = -INFINITY; int bidx = 0;
    for (int m = lane; m < N_; m += 32) {
      const float v = rb[m];
      if (v > best) { best = v; bidx = m; }
    }
    for (int off = 16; off; off >>= 1) {
      const float ov = __shfl_down(best, off, 32);
      const int   oi = __shfl_down(bidx, off, 32);
      if (ov > best) { best = ov; bidx = oi; }
    }
    if (lane == 0) {
      lp[k]        = best;
      e0[eb + k]   = (float)(bidx + b * N_);
      e1[eb + k]   = (float)(n + b * N_);
      rb[bidx]     = -INFINITY;
    }
    __syncthreads();
  }
}

// ---------------------------------------------------------------------------
extern "C" void kernel_launch(void* const* d_in, const int* in_sizes, int n_in,
                              void* d_out, int out_size, void* d_ws, size_t ws_size,
                              hipStream_t stream) {
  (void)in_sizes; (void)n_in; (void)out_size; (void)ws_size;
  const float* x           = (const float*)d_in[0];
  const float* W           = (const float*)d_in[1];
  const float* temperature = (const float*)d_in[2];
  const float* q           = (const float*)d_in[3];
  const int*   A_init      = (const int*)d_in[4];
  float* out = (float*)d_out;

  // workspace layout (floats): xe | x2 | count | prob | m1   (~66 MB)
  float* xe    = (float*)d_ws;
  float* x2    = xe + (size_t)ROWS * D_;
  float* count = x2 + ROWS;
  float* prob  = count + ROWS;
  float* m1    = prob + (size_t)B_ * N_ * N_;

  float* out_x  = out;                               // B*N*D   = 524288
  float* out_e  = out + (size_t)ROWS * D_;           // 2*B*N*K = 131072
  float* out_lp = out_e + (size_t)2 * B_ * N_ * K_;  // B*N*K   = 65536

  embed_kernel    <<<ROWS / 16 / 4,       128, 0, stream>>>(x, W, xe, out_x);
  rownorm_kernel  <<<ROWS * 32 / 256,     256, 0, stream>>>(xe, x2);
  colcount_kernel <<<ROWS / 256,          256, 0, stream>>>(A_init, count);
  gram_prob_kernel<<<(B_ * 128 * 16) / 4, 128, 0, stream>>>(xe, x2, temperature, prob);
  m1_gemm_kernel  <<<(B_ * 128 * 16) / 8, 256, 0, stream>>>(prob, A_init, count, m1);
  merge_topk_kernel<<<ROWS / 2,            64, 0, stream>>>(m1, q, out_e, out_lp);
}